// SS2D_27453430956041
// MI455X (gfx1250) — compile-verified
//
#include <hip/hip_runtime.h>
#include <hip/hip_bf16.h>

typedef float v2f __attribute__((ext_vector_type(2)));
typedef float v8f __attribute__((ext_vector_type(8)));

namespace {
constexpr int kB = 8;
constexpr int kC = 128;   // channels d
constexpr int kH = 96;
constexpr int kW = 96;
constexpr int kL = kH * kW;      // 9216 tokens per image
constexpr int kN = kB * kL;      // 73728 tokens total
constexpr int kXS = 68;          // LDS stride (floats) for x tile [chan][token]
constexpr int kYS = 132;         // LDS stride (floats) for ysum tile [token][chan]
}

__device__ __forceinline__ float silu_f(float x) {
    return x / (1.0f + __expf(-x));
}

#if defined(__gfx1250__) && __has_builtin(__builtin_amdgcn_global_load_async_to_lds_b128)
#define K1_ASYNC_LDS 1
typedef int async_v4i __attribute__((vector_size(4 * sizeof(int))));
typedef __attribute__((address_space(1))) async_v4i* async_gptr;   // global
typedef __attribute__((address_space(3))) async_v4i* async_lptr;   // LDS
#else
#define K1_ASYNC_LDS 0
#endif

// ---------------------------------------------------------------------------
// Kernel 1: uv = x @ W_in^T + b_in  ->  ws_u = silu(uv[:,0:128]), ws_v = uv[:,128:256]
// x layout (B, C, H, W): token t = h*96+w is contiguous for fixed (b, c).
// M-tile = 64 tokens per workgroup; the tile is staged to LDS (async global->LDS
// when available) in chan-major layout; A-fragments are fed from LDS, B from L2.
// fp32 WMMA 16x16x4: A vgpr j holds K = k0 + 2*(lane/16) + j, M = lane%16.
// ---------------------------------------------------------------------------
__global__ __launch_bounds__(256) void k1_gemm_in(
    const float* __restrict__ x,
    const float* __restrict__ W_in,   // (256,128) row-major
    const float* __restrict__ b_in,   // (256)
    float* __restrict__ ws_u,         // (kN,128)
    float* __restrict__ ws_v)         // (kN,128)
{
    __shared__ float lds_x[kC * kXS];          // 128 ch x 64 tok (+pad) = 34.8 KB

    const int t0   = blockIdx.x * 64;          // never crosses image boundary (9216 % 64 == 0)
    const int b    = t0 / kL;
    const int ti   = t0 % kL;
    const int lane = threadIdx.x & 31;
    const int wv   = threadIdx.x >> 5;
    const int hi   = lane >> 4;                // 0/1 lane half
    const int lm   = lane & 15;
    const int n0   = wv * 32;                  // each wave: two 16-wide N tiles

    const float* xb = x + (size_t)b * kC * kL;

    // ---- Stage x tile into LDS: lds_x[c*kXS + tok], tok in [0,64) ----
    {
        const int tid = threadIdx.x;
        #pragma unroll
        for (int i = 0; i < 8; ++i) {
            const int e  = tid + i * 256;      // 2048 16-byte chunks
            const int c  = e >> 4;             // channel 0..127
            const int tq = (e & 15) * 4;       // token quad 0,4,...,60
            const float* gp = xb + (size_t)c * kL + (ti + tq);
            float*       lp = lds_x + c * kXS + tq;
#if K1_ASYNC_LDS
            __builtin_amdgcn_global_load_async_to_lds_b128(
                (async_gptr)(uintptr_t)gp,
                (async_lptr)(unsigned)(uintptr_t)lp,
                0, 0);
#else
            *(float4*)lp = *(const float4*)gp;
#endif
        }
#if K1_ASYNC_LDS
#if __has_builtin(__builtin_amdgcn_s_wait_asynccnt)
        __builtin_amdgcn_s_wait_asynccnt(0);
#else
        asm volatile("s_wait_asynccnt 0x0" ::: "memory");
#endif
#endif
        __syncthreads();
    }

    v8f acc[4][2];
    #pragma unroll
    for (int m = 0; m < 4; ++m)
        #pragma unroll
        for (int j = 0; j < 2; ++j)
            acc[m][j] = (v8f)0.0f;

    for (int kt = 0; kt < 32; ++kt) {
        const int k = kt * 4 + 2 * hi;
        v2f a[4];
        #pragma unroll
        for (int m = 0; m < 4; ++m) {
            const int tok = m * 16 + lm;
            a[m].x = lds_x[(k)     * kXS + tok];   // channel k
            a[m].y = lds_x[(k + 1) * kXS + tok];   // channel k+1
        }
        v2f bf[2];
        #pragma unroll
        for (int j = 0; j < 2; ++j)                // 8B-aligned (k even) -> b64
            bf[j] = *(const v2f*)(W_in + (size_t)(n0 + j * 16 + lm) * kC + k);
        #pragma unroll
        for (int m = 0; m < 4; ++m)
            #pragma unroll
            for (int j = 0; j < 2; ++j)
                acc[m][j] = __builtin_amdgcn_wmma_f32_16x16x4_f32(
                    false, a[m], false, bf[j], (short)0, acc[m][j], false, false);
    }

    // C/D layout: vgpr r -> M = r + 8*(lane/16), N = lane%16 within the tile
    #pragma unroll
    for (int m = 0; m < 4; ++m) {
        #pragma unroll
        for (int j = 0; j < 2; ++j) {
            const int n    = n0 + j * 16 + lm;
            const float bi = b_in[n];
            #pragma unroll
            for (int r = 0; r < 8; ++r) {
                const int tok = t0 + m * 16 + r + 8 * hi;
                const float val = acc[m][j][r] + bi;
                if (n < kC) ws_u[(size_t)tok * kC + n]        = silu_f(val);
                else        ws_v[(size_t)tok * kC + (n - kC)] = val;
            }
        }
    }
}

// ---------------------------------------------------------------------------
// Kernel 2: one workgroup per (b, h) row of 96 pixels.
//  Phase A: 7-tap depthwise conv in lr-order and tb-order, each with the
//           kernel in both orientations (= the two reversed directions mapped
//           back to original positions), SiLU gate, 4x LayerNorm over channels,
//           sum * 0.25 -> LDS.
//  Phase B: one fp32-WMMA GEMM (96x128) @ W_out^T + b_out -> d_out (B,C,H,W).
// ---------------------------------------------------------------------------
__global__ __launch_bounds__(256) void k2_mix_out(
    const float* __restrict__ ws_u,
    const float* __restrict__ ws_v,
    const float* __restrict__ dw_w,   // (128,1,7)
    const float* __restrict__ dw_b,   // (128)
    const float* __restrict__ gamma,  // (128)
    const float* __restrict__ beta,   // (128)
    const float* __restrict__ W_out,  // (128,128)
    const float* __restrict__ b_out,  // (128)
    float* __restrict__ out)          // (B,128,96,96)
{
    __shared__ float ysum[96 * kYS];  // 96 tokens x 128 ch, stride 132 (conflict-free b64)

    const int b    = blockIdx.x / kH;
    const int h    = blockIdx.x % kH;
    const int lane = threadIdx.x & 31;
    const int wv   = threadIdx.x >> 5;
    const int hi   = lane >> 4;
    const int lm   = lane & 15;
    const int c0   = lane * 4;        // each lane owns 4 channels in phase A

    // Per-lane channel constants (fixed across the whole row)
    float dwA[4][7];
    #pragma unroll
    for (int j = 0; j < 4; ++j)
        #pragma unroll
        for (int k = 0; k < 7; ++k)
            dwA[j][k] = dw_w[(c0 + j) * 7 + k];
    const float4 g4  = *(const float4*)(gamma + c0);
    const float4 be4 = *(const float4*)(beta + c0);
    const float4 db4 = *(const float4*)(dw_b + c0);
    const float gg[4] = {g4.x, g4.y, g4.z, g4.w};
    const float bb[4] = {be4.x, be4.y, be4.z, be4.w};
    const float db[4] = {db4.x, db4.y, db4.z, db4.w};

    const size_t imgbase = (size_t)b * kL;

    // -------- Phase A: conv + gate + 4x LayerNorm, summed --------
    for (int w = wv; w < kW; w += 8) {
        const int tlr = h * kW + w;                 // lr-order flat token
        const float4 u4 = *(const float4*)(ws_u + (imgbase + tlr) * kC + c0);
        const float uu[4] = {u4.x, u4.y, u4.z, u4.w};

        float clr[4] = {0,0,0,0}, crl[4] = {0,0,0,0};
        float ctb[4] = {0,0,0,0}, cbt[4] = {0,0,0,0};
        #pragma unroll
        for (int k = 0; k < 7; ++k) {
            const int tt = tlr + k - 3;             // lr sequence neighbor
            if (tt >= 0 && tt < kL) {
                const float4 tp = *(const float4*)(ws_v + (imgbase + tt) * kC + c0);
                const float t[4] = {tp.x, tp.y, tp.z, tp.w};
                #pragma unroll
                for (int j = 0; j < 4; ++j) {
                    clr[j] += dwA[j][k]     * t[j];
                    crl[j] += dwA[j][6 - k] * t[j];   // reversed direction = flipped taps
                }
            }
            const int pp = w * kH + h + k - 3;      // tb sequence neighbor
            if (pp >= 0 && pp < kL) {
                const int h2 = pp % kH, w2 = pp / kH;
                const float4 tp = *(const float4*)(ws_v + (imgbase + h2 * kW + w2) * kC + c0);
                const float t[4] = {tp.x, tp.y, tp.z, tp.w};
                #pragma unroll
                for (int j = 0; j < 4; ++j) {
                    ctb[j] += dwA[j][k]     * t[j];
                    cbt[j] += dwA[j][6 - k] * t[j];
                }
            }
        }

        float y[4][4];                              // [direction][channel]
        #pragma unroll
        for (int j = 0; j < 4; ++j) {
            y[0][j] = silu_f(clr[j] + db[j]) * uu[j];
            y[1][j] = silu_f(crl[j] + db[j]) * uu[j];
            y[2][j] = silu_f(ctb[j] + db[j]) * uu[j];
            y[3][j] = silu_f(cbt[j] + db[j]) * uu[j];
        }

        float s[4], q[4];
        #pragma unroll
        for (int i = 0; i < 4; ++i) {
            s[i] = y[i][0] + y[i][1] + y[i][2] + y[i][3];
            q[i] = y[i][0]*y[i][0] + y[i][1]*y[i][1] + y[i][2]*y[i][2] + y[i][3]*y[i][3];
        }
        #pragma unroll
        for (int m = 16; m >= 1; m >>= 1) {
            #pragma unroll
            for (int i = 0; i < 4; ++i) {
                s[i] += __shfl_xor(s[i], m, 32);
                q[i] += __shfl_xor(q[i], m, 32);
            }
        }
        float mu[4], rs[4];
        #pragma unroll
        for (int i = 0; i < 4; ++i) {
            mu[i] = s[i] * (1.0f / 128.0f);
            float var = q[i] * (1.0f / 128.0f) - mu[i] * mu[i];
            rs[i] = rsqrtf(var + 1e-5f);
        }
        float o[4];
        #pragma unroll
        for (int j = 0; j < 4; ++j) {
            float a = 0.0f;
            #pragma unroll
            for (int i = 0; i < 4; ++i) a += (y[i][j] - mu[i]) * rs[i];
            o[j] = 0.25f * gg[j] * a + bb[j];       // gamma*sum*0.25 + beta (beta*4*0.25)
        }
        float4 ov; ov.x = o[0]; ov.y = o[1]; ov.z = o[2]; ov.w = o[3];
        *(float4*)(ysum + w * kYS + c0) = ov;
    }

    __syncthreads();

    // -------- Phase B: out_row = ysum(96x128) @ W_out^T + b_out --------
    // 6 M-tiles x 8 N-tiles = 48 tiles; 6 per wave.
    for (int tile = wv; tile < 48; tile += 8) {
        const int mt = tile % 6;
        const int nt = tile / 6;
        const int n  = nt * 16 + lm;
        v8f acc = (v8f)0.0f;
        for (int kt = 0; kt < 32; ++kt) {
            const int k = kt * 4 + 2 * hi;
            // both 8B-aligned (k even, kYS even) -> single b64 loads
            const v2f a  = *(const v2f*)(ysum + (mt * 16 + lm) * kYS + k);
            const v2f bf = *(const v2f*)(W_out + (size_t)n * kC + k);
            acc = __builtin_amdgcn_wmma_f32_16x16x4_f32(
                false, a, false, bf, (short)0, acc, false, false);
        }
        const float bias = b_out[n];
        #pragma unroll
        for (int r = 0; r < 8; ++r) {
            const int w_ = mt * 16 + r + 8 * hi;
            out[(((size_t)b * kC + n) * kH + h) * kW + w_] = acc[r] + bias;
        }
    }
}

extern "C" void kernel_launch(void* const* d_in, const int* in_sizes, int n_in,
                              void* d_out, int out_size, void* d_ws, size_t ws_size,
                              hipStream_t stream) {
    const float* x     = (const float*)d_in[0];
    const float* W_in  = (const float*)d_in[1];
    const float* b_in  = (const float*)d_in[2];
    const float* dw_w  = (const float*)d_in[3];
    const float* dw_b  = (const float*)d_in[4];
    const float* gamma = (const float*)d_in[5];
    const float* beta  = (const float*)d_in[6];
    const float* W_out = (const float*)d_in[7];
    const float* b_out = (const float*)d_in[8];
    float* out  = (float*)d_out;

    float* ws_u = (float*)d_ws;                       // kN * 128 floats
    float* ws_v = ws_u + (size_t)kN * kC;             // kN * 128 floats (total 75.5 MB)

    k1_gemm_in<<<kN / 64, 256, 0, stream>>>(x, W_in, b_in, ws_u, ws_v);
    k2_mix_out<<<kB * kH, 256, 0, stream>>>(ws_u, ws_v, dw_w, dw_b, gamma, beta,
                                            W_out, b_out, out);
}